// GConv_5471788335193
// MI455X (gfx1250) — compile-verified
//
#include <hip/hip_runtime.h>

typedef __attribute__((ext_vector_type(2))) float v2f;
typedef __attribute__((ext_vector_type(8))) float v8f;

#define D_FEAT 256

// ---------------- degree / normalization ----------------

__global__ void k_init_deg(float* __restrict__ deg, int N) {
  int n = blockIdx.x * blockDim.x + threadIdx.x;
  if (n < N) deg[n] = 1.0f;  // self-loop contributes 1 to every node's degree
}

__global__ void k_edge_deg(const long long* __restrict__ ei, float* __restrict__ deg, int E) {
  int e = blockIdx.x * blockDim.x + threadIdx.x;
  if (e < E) {
    int dst = (int)ei[(size_t)E + e];
    atomicAdd(&deg[dst], 1.0f);
  }
}

__global__ void k_rsqrt_deg(float* __restrict__ deg, int N) {
  int n = blockIdx.x * blockDim.x + threadIdx.x;
  if (n < N) deg[n] = rsqrtf(deg[n]);  // deg >= 1 always (self-loops)
}

// ---------------- fp32 WMMA GEMM: H[N,256] = X[N,256] * W[256,256] ----------------
// Block: 512 threads = 16 waves. Block stages a 16-row x 256-col fp32 stripe of X
// in LDS; wave w computes the 16x16 output tile at columns [16w, 16w+16).
// K is consumed 4 at a time with V_WMMA_F32_16X16X4_F32 (exact fp32 math).

__global__ __launch_bounds__(512) void k_gemm_f32_wmma(const float* __restrict__ X,
                                                       const float* __restrict__ W,
                                                       float* __restrict__ H) {
  __shared__ float ldsA[16 * D_FEAT];  // 16 KB

  const int t   = threadIdx.x;
  const int m0  = blockIdx.x * 16;

  // Cooperative, fully coalesced stage of 16 contiguous rows of X (16*256 floats).
  {
    const float4* src = (const float4*)(X + (size_t)m0 * D_FEAT);
    float4* dst = (float4*)ldsA;
    dst[t * 2]     = src[t * 2];
    dst[t * 2 + 1] = src[t * 2 + 1];
  }
  __syncthreads();

  const int wave = t >> 5;         // 0..15 -> N tile
  const int lane = t & 31;
  const int n0   = wave * 16;
  const int half = lane >> 4;      // 0: lanes 0-15, 1: lanes 16-31
  const int l16  = lane & 15;

  // ISA 16x4 fp32 A layout: lane(0-15) holds (M=lane, K=k0+0/+1),
  // lane(16-31) holds (M=lane-16, K=k0+2/+3). B (4x16) mirrors on K.
  const int koff = half * 2;
  const int bcol = n0 + l16;

  v8f acc = {};
#pragma unroll 4
  for (int k0 = 0; k0 < D_FEAT; k0 += 4) {
    const int ka = k0 + koff;
    v2f a, b;
    a.x = ldsA[l16 * D_FEAT + ka];
    a.y = ldsA[l16 * D_FEAT + ka + 1];
    b.x = W[(size_t)ka * D_FEAT + bcol];
    b.y = W[(size_t)(ka + 1) * D_FEAT + bcol];
    acc = __builtin_amdgcn_wmma_f32_16x16x4_f32(
        /*neg_a=*/false, a, /*neg_b=*/false, b,
        /*c_mod=*/(short)0, acc, /*reuse_a=*/false, /*reuse_b=*/false);
  }

  // C/D layout: VGPR r -> M = r (lanes 0-15) / r+8 (lanes 16-31), N = lane&15.
  const int rbase = m0 + half * 8;
#pragma unroll
  for (int r = 0; r < 8; ++r) {
    H[(size_t)(rbase + r) * D_FEAT + bcol] = acc[r];
  }
}

// ---------------- aggregation ----------------

// agg[n,:] = h[n,:] * dinv[n]^2   (self-loop message)
__global__ void k_self_init(const float* __restrict__ h, const float* __restrict__ dinv,
                            float* __restrict__ agg, int N) {
  unsigned i = blockIdx.x * blockDim.x + threadIdx.x;  // over N*64 float4 groups
  if (i >= (unsigned)N * 64u) return;
  int n = i >> 6;
  float w = dinv[n] * dinv[n];
  float4 v = ((const float4*)h)[i];
  v.x *= w; v.y *= w; v.z *= w; v.w *= w;
  ((float4*)agg)[i] = v;
}

// agg[dst,:] += h[src,:] * dinv[src]*dinv[dst]   (one thread = one edge x 4 feats)
__global__ void k_scatter(const long long* __restrict__ ei, const float* __restrict__ h,
                          const float* __restrict__ dinv, float* __restrict__ agg, int E) {
  unsigned i = blockIdx.x * blockDim.x + threadIdx.x;  // over E*64
  if (i >= (unsigned)E * 64u) return;
  int e = i >> 6;
  int g = (i & 63u) << 2;
  int s = (int)ei[e];
  int d = (int)ei[(size_t)E + e];
  float w = dinv[s] * dinv[d];
  float4 m = *(const float4*)(h + (size_t)s * D_FEAT + g);
  float* ap = agg + (size_t)d * D_FEAT + g;
  atomicAdd(ap + 0, m.x * w);
  atomicAdd(ap + 1, m.y * w);
  atomicAdd(ap + 2, m.z * w);
  atomicAdd(ap + 3, m.w * w);
}

// out[n,:] = relu(agg[n,:] + b[:])   (safe for agg == out, element-wise)
__global__ void k_finalize(const float* __restrict__ agg, const float* __restrict__ bias,
                           float* __restrict__ out, int N) {
  unsigned i = blockIdx.x * blockDim.x + threadIdx.x;  // over N*64
  if (i >= (unsigned)N * 64u) return;
  int g = (i & 63u) << 2;
  float4 v = ((const float4*)agg)[i];
  float4 b = *(const float4*)(bias + g);
  v.x = fmaxf(v.x + b.x, 0.0f);
  v.y = fmaxf(v.y + b.y, 0.0f);
  v.z = fmaxf(v.z + b.z, 0.0f);
  v.w = fmaxf(v.w + b.w, 0.0f);
  ((float4*)out)[i] = v;
}

// ---------------- driver ----------------

extern "C" void kernel_launch(void* const* d_in, const int* in_sizes, int n_in,
                              void* d_out, int out_size, void* d_ws, size_t ws_size,
                              hipStream_t stream) {
  const float*     x  = (const float*)d_in[0];
  const long long* ei = (const long long*)d_in[1];  // int64 edge_index [2, E]
  const float*     W1 = (const float*)d_in[2];
  const float*     b1 = (const float*)d_in[3];
  const float*     W2 = (const float*)d_in[4];
  const float*     b2 = (const float*)d_in[5];
  float*           out = (float*)d_out;

  const int N = in_sizes[0] / D_FEAT;  // 50000
  const int E = in_sizes[1] / 2;       // 1600000

  // Workspace: dinv[N] (padded), h[N*256]. d_out is reused as the agg buffer.
  float* dinv = (float*)d_ws;
  size_t Na = ((size_t)N + 63) & ~(size_t)63;
  float* h = dinv + Na;

  const int TPB = 256;
  dim3 gN((N + TPB - 1) / TPB);
  dim3 gE((E + TPB - 1) / TPB);
  dim3 gNF(((unsigned)N * 64u + TPB - 1) / TPB);
  dim3 gEF(((unsigned)E * 64u + TPB - 1) / TPB);
  dim3 gGemm(N / 16);  // 3125 (N divisible by 16)

  // symmetric normalization dinv = rsqrt(deg)
  k_init_deg <<<gN, TPB, 0, stream>>>(dinv, N);
  k_edge_deg <<<gE, TPB, 0, stream>>>(ei, dinv, E);
  k_rsqrt_deg<<<gN, TPB, 0, stream>>>(dinv, N);

  // ---- layer 1 ----
  k_gemm_f32_wmma<<<gGemm, 512, 0, stream>>>(x, W1, h);
  k_self_init    <<<gNF, TPB, 0, stream>>>(h, dinv, out, N);   // agg in d_out
  k_scatter      <<<gEF, TPB, 0, stream>>>(ei, h, dinv, out, E);
  k_finalize     <<<gNF, TPB, 0, stream>>>(out, b1, out, N);   // out = z1

  // ---- layer 2 ----
  k_gemm_f32_wmma<<<gGemm, 512, 0, stream>>>(out, W2, h);      // reads z1
  k_self_init    <<<gNF, TPB, 0, stream>>>(h, dinv, out, N);   // agg overwrites z1
  k_scatter      <<<gEF, TPB, 0, stream>>>(ei, h, dinv, out, E);
  k_finalize     <<<gNF, TPB, 0, stream>>>(out, b2, out, N);
}